// Attn_12360915878014
// MI455X (gfx1250) — compile-verified
//
#include <hip/hip_runtime.h>

typedef __attribute__((ext_vector_type(2))) float v2f;
typedef __attribute__((ext_vector_type(8))) float v8f;

#define HDIM 1024
#define SDIM 32768

// ---------------------------------------------------------------------------
// GEMV y[0..rows) = M (rows x HDIM, row-major) * x + bias, via f32 WMMA.
// One wave computes 16 rows. Block = 256 threads = 8 waves = 128 rows.
// ---------------------------------------------------------------------------
__global__ __launch_bounds__(256) void gemv_wmma_bias(
    const float* __restrict__ M, const float* __restrict__ x,
    const float* __restrict__ bias, float* __restrict__ y) {
  __shared__ float xs[HDIM];
  for (int i = threadIdx.x; i < HDIM; i += 256) xs[i] = x[i];
  __syncthreads();

  const int lane    = threadIdx.x & 31;
  const int wave    = threadIdx.x >> 5;
  const int half    = lane >> 4;      // K half: lanes 0-15 -> K{0,1}, 16-31 -> K{2,3}
  const int m       = lane & 15;      // A-matrix row within tile
  const int rowBase = (blockIdx.x * 8 + wave) * 16;

  const float* rowPtr = M + (size_t)(rowBase + m) * HDIM + (half << 1);
  const float* xp     = xs + (half << 1);

  v8f c0 = {}; v8f c1 = {};
  #pragma unroll 4
  for (int k = 0; k < HDIM; k += 8) {
    v2f a0 = *(const v2f*)(rowPtr + k);
    v2f b0 = *(const v2f*)(xp + k);
    v2f a1 = *(const v2f*)(rowPtr + k + 4);
    v2f b1 = *(const v2f*)(xp + k + 4);
    c0 = __builtin_amdgcn_wmma_f32_16x16x4_f32(false, a0, false, b0,
                                               (short)0, c0, false, false);
    c1 = __builtin_amdgcn_wmma_f32_16x16x4_f32(false, a1, false, b1,
                                               (short)0, c1, false, false);
  }
  v8f c = c0 + c1;  // all D columns identical (B columns identical)

  if (m == 0) {  // lane 0 -> rows 0..7, lane 16 -> rows 8..15
    const int r0 = rowBase + (half << 3);
    #pragma unroll
    for (int r = 0; r < 8; ++r) y[r0 + r] = c[r] + bias[r0 + r];
  }
}

// ---------------------------------------------------------------------------
// Big GEMV: e[0..SDIM) = enc (SDIM x HDIM) * energy, + per-block max.
// 256 blocks x 8 waves x 16 rows = 32768 rows. Streaming-bandwidth bound.
// ---------------------------------------------------------------------------
__global__ __launch_bounds__(256) void gemv_wmma_emax(
    const float* __restrict__ Mat, const float* __restrict__ x,
    float* __restrict__ e, float* __restrict__ blockMax) {
  __shared__ float xs[HDIM];
  __shared__ float wmax[16];
  for (int i = threadIdx.x; i < HDIM; i += 256) xs[i] = x[i];
  __syncthreads();

  const int lane    = threadIdx.x & 31;
  const int wave    = threadIdx.x >> 5;
  const int half    = lane >> 4;
  const int m       = lane & 15;
  const int rowBase = (blockIdx.x * 8 + wave) * 16;

  const float* rowPtr = Mat + (size_t)(rowBase + m) * HDIM + (half << 1);
  const float* xp     = xs + (half << 1);

  v8f c0 = {}; v8f c1 = {};
  #pragma unroll 4
  for (int k = 0; k < HDIM; k += 8) {
    __builtin_prefetch(rowPtr + k + 128, 0, 0);  // global_prefetch_b8, ~512B ahead
    v2f a0 = *(const v2f*)(rowPtr + k);
    v2f b0 = *(const v2f*)(xp + k);
    v2f a1 = *(const v2f*)(rowPtr + k + 4);
    v2f b1 = *(const v2f*)(xp + k + 4);
    c0 = __builtin_amdgcn_wmma_f32_16x16x4_f32(false, a0, false, b0,
                                               (short)0, c0, false, false);
    c1 = __builtin_amdgcn_wmma_f32_16x16x4_f32(false, a1, false, b1,
                                               (short)0, c1, false, false);
  }
  v8f c = c0 + c1;

  if (m == 0) {
    const int r0 = rowBase + (half << 3);
    float mx = -3.402823466e38f;
    #pragma unroll
    for (int r = 0; r < 8; ++r) {
      float v = c[r];
      e[r0 + r] = v;
      mx = fmaxf(mx, v);
    }
    wmax[(wave << 1) | half] = mx;
  }
  __syncthreads();
  if (threadIdx.x == 0) {
    float mx = wmax[0];
    #pragma unroll
    for (int i = 1; i < 16; ++i) mx = fmaxf(mx, wmax[i]);
    blockMax[blockIdx.x] = mx;
  }
}

// ---------------------------------------------------------------------------
// Softmax reductions (deterministic tree reductions, no float atomics)
// ---------------------------------------------------------------------------
__global__ __launch_bounds__(256) void reduce_max256(
    const float* __restrict__ in, float* __restrict__ gmax) {
  __shared__ float s[256];
  s[threadIdx.x] = in[threadIdx.x];
  __syncthreads();
  for (int st = 128; st > 0; st >>= 1) {
    if (threadIdx.x < st) s[threadIdx.x] = fmaxf(s[threadIdx.x], s[threadIdx.x + st]);
    __syncthreads();
  }
  if (threadIdx.x == 0) *gmax = s[0];
}

__global__ __launch_bounds__(256) void exp_partial(
    const float* __restrict__ e, const float* __restrict__ gmax,
    float* __restrict__ p, float* __restrict__ blockSum) {
  __shared__ float s[256];
  const int i = blockIdx.x * 256 + threadIdx.x;
  float v = expf(e[i] - *gmax);
  p[i] = v;
  s[threadIdx.x] = v;
  __syncthreads();
  for (int st = 128; st > 0; st >>= 1) {
    if (threadIdx.x < st) s[threadIdx.x] += s[threadIdx.x + st];
    __syncthreads();
  }
  if (threadIdx.x == 0) blockSum[blockIdx.x] = s[0];
}

__global__ __launch_bounds__(128) void reduce_sum_inv128(
    const float* __restrict__ in, float* __restrict__ ginv) {
  __shared__ float s[128];
  s[threadIdx.x] = in[threadIdx.x];
  __syncthreads();
  for (int st = 64; st > 0; st >>= 1) {
    if (threadIdx.x < st) s[threadIdx.x] += s[threadIdx.x + st];
    __syncthreads();
  }
  if (threadIdx.x == 0) *ginv = 1.0f / s[0];
}

__global__ __launch_bounds__(256) void scale_out(
    float* __restrict__ p, const float* __restrict__ ginv) {
  const int i = blockIdx.x * 256 + threadIdx.x;
  p[i] *= *ginv;
}

// ---------------------------------------------------------------------------
extern "C" void kernel_launch(void* const* d_in, const int* in_sizes, int n_in,
                              void* d_out, int out_size, void* d_ws, size_t ws_size,
                              hipStream_t stream) {
  const float* hidden = (const float*)d_in[0];
  const float* enc    = (const float*)d_in[1];
  const float* W_lin  = (const float*)d_in[2];
  const float* b_lin  = (const float*)d_in[3];
  const float* W_attn = (const float*)d_in[4];
  const float* b_attn = (const float*)d_in[5];
  float* out = (float*)d_out;

  float* ws     = (float*)d_ws;
  float* h      = ws;                   // 1024
  float* energy = ws + HDIM;            // 1024
  float* e      = ws + 2 * HDIM;        // 32768
  float* bmax   = e + SDIM;             // 256
  float* gmax   = bmax + 256;           // 1
  float* bsum   = gmax + 1;             // 128
  float* ginv   = bsum + 128;           // 1

  // h = W_lin * hidden + b_lin         (64 waves)
  gemv_wmma_bias<<<8, 256, 0, stream>>>(W_lin, hidden, b_lin, h);
  // energy = W_attn * h + b_attn       (64 waves)
  gemv_wmma_bias<<<8, 256, 0, stream>>>(W_attn, h, b_attn, energy);
  // e = enc * energy, block maxes      (2048 waves, 128 MiB stream)
  gemv_wmma_emax<<<256, 256, 0, stream>>>(enc, energy, e, bmax);
  // softmax
  reduce_max256<<<1, 256, 0, stream>>>(bmax, gmax);
  exp_partial<<<128, 256, 0, stream>>>(e, gmax, out, bsum);
  reduce_sum_inv128<<<1, 128, 0, stream>>>(bsum, ginv);
  scale_out<<<128, 256, 0, stream>>>(out, ginv);
}